// FastFlowDecoder_6657199309274
// MI455X (gfx1250) — compile-verified
//
#include <hip/hip_runtime.h>

typedef __attribute__((ext_vector_type(2))) float v2f;
typedef __attribute__((ext_vector_type(8))) float v8f;

#define BDIM  64            // 2 waves per block (wave32)
#define WPB   2
#define KF    131           // real feature count
#define KP4   132           // padded K: 33 chunks of 4
#define FSTR  134           // f32 LDS row stride (rows 8B-aligned, distinct banks for 16 rows)
#define HSTR  36
#define CCH   64
#define HH    512
#define WW    512
#define HWIMG (HH*WW)
#define NPTS  150000
#define NB    2
#define TILES ((NB*NPTS)/16)   // 18750 exact

#if defined(__gfx1250__) && __has_builtin(__builtin_amdgcn_global_load_async_to_lds_b32)
#define HAVE_ASYNC_LDS 1
#else
#define HAVE_ASYNC_LDS 0
#endif

typedef __attribute__((address_space(1))) int glb_i32;
typedef __attribute__((address_space(3))) int lds_i32;

__device__ __forceinline__ void gather_one(const float* gp, float* lp) {
#if HAVE_ASYNC_LDS
    // scattered global -> LDS, no VGPR writeback, tracked by ASYNCcnt
    int* gi = (int*)const_cast<float*>(gp);   // generic, reinterpreted
    int* li = (int*)lp;
    __builtin_amdgcn_global_load_async_to_lds_b32(
        (glb_i32*)gi, (lds_i32*)li, 0, 0);
#else
    *lp = *gp;   // sync fallback (load + ds_store)
#endif
}

__global__ __launch_bounds__(BDIM)
void fastflow_decoder_kernel(const float* __restrict__ before,
                             const float* __restrict__ after,
                             const float* __restrict__ offs,
                             const int*   __restrict__ vc,
                             const float* __restrict__ W1,
                             const float* __restrict__ b1,
                             const float* __restrict__ W2,
                             const float* __restrict__ b2,
                             float*       __restrict__ out)
{
    __shared__ __align__(16) float sW1t[32 * FSTR];        // W1^T f32: [col n][k]
    __shared__ __align__(16) float sFeat[WPB][16 * FSTR];  // per-wave feats f32: [point][k]
    __shared__ __align__(16) float sH[WPB][16 * HSTR];     // per-wave hidden
    __shared__              float  sW2[96];

    const int tid  = threadIdx.x;
    const int wave = tid >> 5;
    const int lane = tid & 31;

    // ---- one-time per-block staging of W1^T (f32, zero-padded to KP4) and W2 ----
    for (int idx = tid; idx < 32 * KP4; idx += BDIM) {
        int k = idx >> 5, nn = idx & 31;
        sW1t[nn * FSTR + k] = (k < KF) ? W1[k * 32 + nn] : 0.0f;
    }
    for (int idx = tid; idx < 96; idx += BDIM) sW2[idx] = W2[idx];
    __syncthreads();

    const int pl    = lane & 15;        // point-in-tile for this lane
    const int khalf = lane >> 4;        // 0: lanes 0-15 (K 0,1), 1: lanes 16-31 (K 2,3)

    float* sF  = sFeat[wave];
    float* sHw = sH[wave];

    for (int tile = blockIdx.x * WPB + wave; tile < TILES; tile += gridDim.x * WPB) {
        const int g0 = tile * 16;
        const int b  = g0 / NPTS;
        const int n0 = g0 - b * NPTS;

        const long pg = (long)b * NPTS + n0 + pl;
        const int  y  = vc[pg * 3 + 1];
        const int  x  = vc[pg * 3 + 2];
        const long poff = (long)y * WW + x;
        const float* imgB = before + (size_t)b * CCH * HWIMG;
        const float* imgA = after  + (size_t)b * CCH * HWIMG;

        // zero the K pad element (k = 131) of this point's row
        if (khalf == 0) sF[pl * FSTR + 131] = 0.0f;

        // ---- scattered gather straight into LDS; lanes 0-15 even k, 16-31 odd k ----
        for (int k = khalf; k < KF; k += 2) {
            const float* gp;
            if (k < 64)       gp = imgB + (size_t)k * HWIMG + poff;
            else if (k < 128) gp = imgA + (size_t)(k - 64) * HWIMG + poff;
            else              gp = offs + pg * 3 + (k - 128);
            gather_one(gp, &sF[pl * FSTR + k]);
        }
#if HAVE_ASYNC_LDS
        asm volatile("s_wait_asynccnt 0x0" ::: "memory");
#endif
        asm volatile("s_wait_dscnt 0x0" ::: "memory");

        // ---- h = feats @ W1 via v_wmma_f32_16x16x4_f32, 33 K-chunks of 4 ----
        // A 16x4 f32: lanes0-15 M=lane K={0,1}; lanes16-31 K={2,3}
        // B 4x16 f32: lanes0-15 N=lane K={0,1}; lanes16-31 K={2,3}
        v8f a0e = {}, a0o = {}, a1e = {}, a1o = {};   // even/odd partial chains
        const float* fRow  = &sF[pl * FSTR + 2 * khalf];
        const float* bRow0 = &sW1t[pl * FSTR + 2 * khalf];          // hidden cols 0..15
        const float* bRow1 = &sW1t[(16 + pl) * FSTR + 2 * khalf];   // hidden cols 16..31
        #pragma unroll
        for (int kc = 0; kc < 33; ++kc) {
            v2f aF  = *(const v2f*)(fRow  + kc * 4);
            v2f bF0 = *(const v2f*)(bRow0 + kc * 4);
            v2f bF1 = *(const v2f*)(bRow1 + kc * 4);
            if (kc & 1) {
                a0o = __builtin_amdgcn_wmma_f32_16x16x4_f32(false, aF, false, bF0, (short)0, a0o, false, false);
                a1o = __builtin_amdgcn_wmma_f32_16x16x4_f32(false, aF, false, bF1, (short)0, a1o, false, false);
            } else {
                a0e = __builtin_amdgcn_wmma_f32_16x16x4_f32(false, aF, false, bF0, (short)0, a0e, false, false);
                a1e = __builtin_amdgcn_wmma_f32_16x16x4_f32(false, aF, false, bF1, (short)0, a1e, false, false);
            }
        }
        v8f acc0 = a0e + a0o;
        v8f acc1 = a1e + a1o;

        // ---- bias + exact GELU, scatter h tile into LDS ----
        // D layout: lanes0-15 -> M=r, lanes16-31 -> M=8+r; N = lane&15 (+16 for acc1)
        const int   c0   = pl;
        const int   c1   = 16 + pl;
        const float bb0  = b1[c0];
        const float bb1  = b1[c1];
        const int   mrow = khalf ? 8 : 0;
        #pragma unroll
        for (int r = 0; r < 8; ++r) {
            float h0 = acc0[r] + bb0;
            h0 = 0.5f * h0 * (1.0f + erff(h0 * 0.70710678118654752f));
            sHw[(mrow + r) * HSTR + c0] = h0;
            float h1 = acc1[r] + bb1;
            h1 = 0.5f * h1 * (1.0f + erff(h1 * 0.70710678118654752f));
            sHw[(mrow + r) * HSTR + c1] = h1;
        }
        asm volatile("s_wait_dscnt 0x0" ::: "memory");

        // ---- layer 2: [16,32] @ [32,3] + b2, one point per lane (lanes 0-15) ----
        if (lane < 16) {
            const long gp = g0 + lane;
            float f0 = b2[0], f1 = b2[1], f2 = b2[2];
            #pragma unroll
            for (int k = 0; k < 32; ++k) {
                float hv = sHw[lane * HSTR + k];
                f0 += hv * sW2[k * 3 + 0];
                f1 += hv * sW2[k * 3 + 1];
                f2 += hv * sW2[k * 3 + 2];
            }
            out[gp * 3 + 0] = f0;
            out[gp * 3 + 1] = f1;
            out[gp * 3 + 2] = f2;
        }
    }
}

extern "C" void kernel_launch(void* const* d_in, const int* in_sizes, int n_in,
                              void* d_out, int out_size, void* d_ws, size_t ws_size,
                              hipStream_t stream) {
    (void)in_sizes; (void)n_in; (void)out_size; (void)d_ws; (void)ws_size;
    const float* before = (const float*)d_in[0];
    const float* after  = (const float*)d_in[1];
    const float* offs   = (const float*)d_in[2];
    const int*   vc     = (const int*)d_in[3];
    const float* W1     = (const float*)d_in[4];
    const float* b1     = (const float*)d_in[5];
    const float* W2     = (const float*)d_in[6];
    const float* b2     = (const float*)d_in[7];
    float* out = (float*)d_out;

    dim3 grid(2048), block(BDIM);   // 4096 waves grid-striding over 18750 tiles
    hipLaunchKernelGGL(fastflow_decoder_kernel, grid, block, 0, stream,
                       before, after, offs, vc, W1, b1, W2, b2, out);
}